// cLN_12197707121207
// MI455X (gfx1250) — compile-verified
//
#include <hip/hip_runtime.h>
#include <math.h>

// Problem constants from the reference setup
#define B_  16
#define C_  256
#define T_  16000
#define EPS 1e-8f

typedef __attribute__((ext_vector_type(2))) float v2f;
typedef __attribute__((ext_vector_type(4))) float f4;
typedef __attribute__((ext_vector_type(8))) float v8f;

// ---------------------------------------------------------------------------
// Pass 1: channel reduction via V_WMMA_F32_16X16X4_F32 (A = ones).
// One wave handles a 16-wide time tile of one batch; 64 K-steps cover C=256.
// accS row0 = sum_c x[c,t], accP row0 = sum_c x[c,t]^2, full f32 precision.
// ---------------------------------------------------------------------------
__global__ __launch_bounds__(256) void cln_reduce_wmma(
    const float* __restrict__ x,
    float* __restrict__ sums,
    float* __restrict__ pows)
{
    const int lane = threadIdx.x & 31;
    const int wave = threadIdx.x >> 5;
    const int tile = blockIdx.x * 8 + wave;       // 16000 tiles total
    const int b    = tile / (T_ / 16);
    const int t0   = (tile % (T_ / 16)) * 16;

    // B-matrix (4x16 KxN) lane mapping: N = lane%16, K = 2*(lane>=16) + vgpr
    const int n  = lane & 15;
    const int kh = (lane >> 4) << 1;              // 0 or 2

    const float* base = x + (size_t)b * C_ * T_ + (size_t)t0 + n;

    v2f a; a[0] = 1.0f; a[1] = 1.0f;              // all-ones A matrix
    v8f accS = {};                                // D = sum
    v8f accP = {};                                // D = power sum

    for (int c = 0; c < C_; c += 4) {
        const float b0 = base[(size_t)(c + kh + 0) * T_];
        const float b1 = base[(size_t)(c + kh + 1) * T_];
        v2f bv; bv[0] = b0;      bv[1] = b1;
        v2f bq; bq[0] = b0 * b0; bq[1] = b1 * b1;
        // (neg_a, A, neg_b, B, c_mod, C, reuse_a, reuse_b)
        accS = __builtin_amdgcn_wmma_f32_16x16x4_f32(
            false, a, false, bv, (short)0, accS, false, false);
        accP = __builtin_amdgcn_wmma_f32_16x16x4_f32(
            false, a, false, bq, (short)0, accP, false, false);
    }

    // C/D layout: VGPR0, lanes 0-15 hold M=0, N=lane. All D rows identical.
    if (lane < 16) {
        sums[b * T_ + t0 + lane] = accS[0];
        pows[b * T_ + t0 + lane] = accP[0];
    }
}

// ---------------------------------------------------------------------------
// Pass 2: per-batch inclusive scan over T of (sum, powsum), then convert to
// (mean, inv_std). One 256-thread block per batch; Hillis-Steele in LDS with
// a uniform register carry across 256-element chunks.
// ---------------------------------------------------------------------------
__global__ __launch_bounds__(256) void cln_scan(
    const float* __restrict__ sums,
    const float* __restrict__ pows,
    float* __restrict__ mean,
    float* __restrict__ istd)
{
    __shared__ float bx[256];
    __shared__ float by[256];

    const int b   = blockIdx.x;
    const int tid = threadIdx.x;

    float cx = 0.0f, cy = 0.0f;                   // running carry (uniform)
    const int nchunks = (T_ + 255) / 256;

    for (int ch = 0; ch < nchunks; ++ch) {
        const int idx = ch * 256 + tid;
        float vx = 0.0f, vy = 0.0f;
        if (idx < T_) {
            vx = sums[b * T_ + idx];
            vy = pows[b * T_ + idx];
        }
        bx[tid] = vx; by[tid] = vy;
        __syncthreads();

        for (int off = 1; off < 256; off <<= 1) {
            float ax = 0.0f, ay = 0.0f;
            if (tid >= off) { ax = bx[tid - off]; ay = by[tid - off]; }
            __syncthreads();
            if (tid >= off) { bx[tid] += ax; by[tid] += ay; }
            __syncthreads();
        }

        const float ix = bx[tid] + cx;            // inclusive prefix + carry
        const float iy = by[tid] + cy;
        if (idx < T_) {
            const float cnt = (float)(idx + 1) * (float)C_;
            const float m   = ix / cnt;
            const float var = (iy - 2.0f * m * ix) / cnt + m * m;
            mean[b * T_ + idx] = m;
            istd[b * T_ + idx] = 1.0f / sqrtf(var + EPS);
        }
        cx += bx[255]; cy += by[255];
        __syncthreads();                          // before buffers are reused
    }
}

// ---------------------------------------------------------------------------
// Pass 3: streaming normalize. float4 along T; NT hints since the 262 MB
// streams exceed the 192 MB L2.
// ---------------------------------------------------------------------------
__global__ __launch_bounds__(256) void cln_norm(
    const float* __restrict__ x,
    const float* __restrict__ mean,
    const float* __restrict__ istd,
    const float* __restrict__ gain,
    const float* __restrict__ bias,
    float* __restrict__ out)
{
    const int T4 = T_ / 4;
    const long long total  = (long long)B_ * C_ * T4;
    const long long stride = (long long)gridDim.x * blockDim.x;
    long long i = (long long)blockIdx.x * blockDim.x + threadIdx.x;

    const f4* xv = (const f4*)x;
    const f4* mv = (const f4*)mean;
    const f4* sv = (const f4*)istd;
    f4*       ov = (f4*)out;

    for (; i < total; i += stride) {
        const int row = (int)(i / T4);
        const int t4  = (int)(i - (long long)row * T4);
        const int b   = row >> 8;                 // row / 256
        const int c   = row & 255;                // row % 256

        const f4 xx = __builtin_nontemporal_load(&xv[i]);
        const f4 m  = mv[b * T4 + t4];
        const f4 s  = sv[b * T4 + t4];
        const float g  = gain[c];
        const float bb = bias[c];

        f4 o = (xx - m) * s * g + bb;
        __builtin_nontemporal_store(o, &ov[i]);
    }
}

// ---------------------------------------------------------------------------
extern "C" void kernel_launch(void* const* d_in, const int* in_sizes, int n_in,
                              void* d_out, int out_size, void* d_ws, size_t ws_size,
                              hipStream_t stream)
{
    (void)in_sizes; (void)n_in; (void)out_size; (void)ws_size;

    const float* x    = (const float*)d_in[0];   // [B, C, T]
    const float* gain = (const float*)d_in[1];   // [1, C, 1]
    const float* bias = (const float*)d_in[2];   // [1, C, 1]
    float*       out  = (float*)d_out;

    float* ws   = (float*)d_ws;                  // 4 * B*T floats = 4 MB
    float* sums = ws;
    float* pows = ws + (size_t)B_ * T_;
    float* mean = ws + (size_t)2 * B_ * T_;
    float* istd = ws + (size_t)3 * B_ * T_;

    // Pass 1: 16000 wave-tiles, 8 waves (256 threads) per block -> 2000 blocks
    cln_reduce_wmma<<<(B_ * (T_ / 16)) / 8, 256, 0, stream>>>(x, sums, pows);

    // Pass 2: one block per batch
    cln_scan<<<B_, 256, 0, stream>>>(sums, pows, mean, istd);

    // Pass 3: one thread per float4
    const int total4 = B_ * C_ * (T_ / 4);       // 16,384,000
    cln_norm<<<total4 / 256, 256, 0, stream>>>(x, mean, istd, gain, bias, out);
}